// SHAM_46557445489415
// MI455X (gfx1250) — compile-verified
//
#include <hip/hip_runtime.h>

// Problem dims (fixed by setup_inputs)
#define B_  16
#define NA  2048
#define NP  2048
#define DD  512
#define KK  32

typedef __attribute__((ext_vector_type(2))) float v2f;
typedef __attribute__((ext_vector_type(8))) float v8f;

// Order-preserving float -> u32 key (total order, works for negatives)
__device__ __forceinline__ unsigned ordkey(float f) {
    unsigned u = __float_as_uint(f);
    return (u & 0x80000000u) ? ~u : (u | 0x80000000u);
}
__device__ __forceinline__ float keyval(unsigned k) {
    return __uint_as_float((k & 0x80000000u) ? (k ^ 0x80000000u) : ~k);
}

// ---------------------------------------------------------------------------
// 1) Gather + L2-normalize the needed anchor rows.
//    rows [0,512): idx_a -> A_s (d_out), rows [512,1024): idx_p -> Asel (ws)
// ---------------------------------------------------------------------------
__global__ __launch_bounds__(256) void prep_kernel(
    const float* __restrict__ anchor, const int* __restrict__ idx_a,
    const int* __restrict__ idx_p, float* __restrict__ a_s_out,
    float* __restrict__ asel_ws) {
    const int r    = blockIdx.x * 8 + (threadIdx.x >> 5);
    const int lane = threadIdx.x & 31;
    const int kind = r >> 9;        // 0: idx_a path, 1: idx_p path
    const int t    = r & 511;       // b*K + k
    const int b    = t >> 5;
    const int idx  = kind ? idx_p[t] : idx_a[t];
    const float* src = anchor + ((size_t)b * NA + idx) * DD;
    float v[16];
    float s = 0.f;
#pragma unroll
    for (int j = 0; j < 16; ++j) { float x = src[lane + 32 * j]; v[j] = x; s += x * x; }
#pragma unroll
    for (int off = 16; off; off >>= 1) s += __shfl_xor(s, off, 32);
    const float inv = 1.f / fmaxf(sqrtf(s), 1e-12f);
    float* dst = (kind ? asel_ws : a_s_out) + (size_t)t * DD;
#pragma unroll
    for (int j = 0; j < 16; ++j) dst[lane + 32 * j] = v[j] * inv;
}

// ---------------------------------------------------------------------------
// 2) Zero the packed (key<<32 | revidx) argmax accumulators
// ---------------------------------------------------------------------------
__global__ void init_best(unsigned long long* best) {
    best[blockIdx.x * 256 + threadIdx.x] = 0ull;
}

// ---------------------------------------------------------------------------
// 3) WMMA scoring: S[32 sel-rows x 128 p-rows] per block, fused row-norm of p,
//    fused argmax via shuffle reduce + global_atomic_max_u64.
//    grid = (NP/128, B), block = 256 (8 waves)
// ---------------------------------------------------------------------------
__global__ __launch_bounds__(256) void score_kernel(
    const float* __restrict__ p, const float* __restrict__ asel,
    unsigned long long* __restrict__ best) {
    __shared__ float As[KK * DD];   // 32 x 512 f32 = 64 KB
    const int b   = blockIdx.y;
    const int tid = threadIdx.x;
    {   // stage Asel[b] into LDS with b128 traffic
        const float4* s4 = (const float4*)(asel + (size_t)b * KK * DD);
        float4* d4 = (float4*)As;
        for (int i = tid; i < KK * DD / 4; i += 256) d4[i] = s4[i];
    }
    __syncthreads();

    const int wave = tid >> 5;
    const int lane = tid & 31;
    const int half = lane >> 4;     // which 16-lane half
    const int lr   = lane & 15;
    const int m    = blockIdx.x * 128 + wave * 16 + lr;   // p-row (N column)

    // ISA f32 fragment layouts: lanes 0-15 carry K0/K1, lanes 16-31 carry K2/K3
    const float* pb = p + ((size_t)b * NP + m) * DD + 2 * half;  // B: 4x16 (KxN)
    const float* a0 = As + lr * DD + 2 * half;                   // A rows 0..15
    const float* a1 = As + (lr + 16) * DD + 2 * half;            // A rows 16..31

    v8f acc0 = {};
    v8f acc1 = {};
    float ss = 0.f;                 // partial ||p_m||^2 (this lane's K-slice)
    for (int k0 = 0; k0 < DD; k0 += 4) {
        v2f bf  = *(const v2f*)(pb + k0);
        v2f af0 = *(const v2f*)(a0 + k0);
        v2f af1 = *(const v2f*)(a1 + k0);
        ss += bf.x * bf.x + bf.y * bf.y;
        acc0 = __builtin_amdgcn_wmma_f32_16x16x4_f32(false, af0, false, bf,
                                                     (short)0, acc0, false, false);
        acc1 = __builtin_amdgcn_wmma_f32_16x16x4_f32(false, af1, false, bf,
                                                     (short)0, acc1, false, false);
    }
    // lanes l and l+16 together cover the full row -> combine halves
    ss += __shfl_xor(ss, 16, 32);
    const float inv = 1.f / fmaxf(sqrtf(ss), 1e-12f);
    const unsigned keylo = (unsigned)(NP - 1 - m);  // invert so max picks lowest idx

#pragma unroll
    for (int i = 0; i < 8; ++i) {
        // C/D layout: VGPR i holds row i (lanes 0-15) and row i+8 (lanes 16-31)
        unsigned long long k0p =
            ((unsigned long long)ordkey(acc0[i] * inv) << 32) | keylo;
        unsigned long long k1p =
            ((unsigned long long)ordkey(acc1[i] * inv) << 32) | keylo;
#pragma unroll
        for (int off = 8; off; off >>= 1) {     // reduce within each 16-lane half
            unsigned long long o0 = __shfl_xor(k0p, off, 32);
            unsigned long long o1 = __shfl_xor(k1p, off, 32);
            k0p = o0 > k0p ? o0 : k0p;
            k1p = o1 > k1p ? o1 : k1p;
        }
        if (lr == 0) {
            atomicMax(best + (size_t)b * KK + (i + 8 * half), k0p);
            atomicMax(best + (size_t)b * KK + (16 + i + 8 * half), k1p);
        }
    }
}

// ---------------------------------------------------------------------------
// 4) Decode winners, weighted-fuse the two matched (normalized) rows, renorm.
//    One wave per (b,k); grid = 64, block = 256
// ---------------------------------------------------------------------------
__global__ __launch_bounds__(256) void fuse_kernel(
    const float* __restrict__ p1, const float* __restrict__ p2,
    const unsigned long long* __restrict__ best, float* __restrict__ p_s_out) {
    const int t    = blockIdx.x * 8 + (threadIdx.x >> 5);
    const int lane = threadIdx.x & 31;
    const int b    = t >> 5;
    const unsigned long long e1 = best[t];
    const unsigned long long e2 = best[B_ * KK + t];
    const float w1 = keyval((unsigned)(e1 >> 32));
    const float w2 = keyval((unsigned)(e2 >> 32));
    const int  i1  = NP - 1 - (int)(unsigned)(e1 & 0xffffffffull);
    const int  i2  = NP - 1 - (int)(unsigned)(e2 & 0xffffffffull);
    const float* r1 = p1 + ((size_t)b * NP + i1) * DD;
    const float* r2 = p2 + ((size_t)b * NP + i2) * DD;

    float v1[16], v2[16];
    float s1 = 0.f, s2 = 0.f;
#pragma unroll
    for (int j = 0; j < 16; ++j) {
        float x = r1[lane + 32 * j]; v1[j] = x; s1 += x * x;
        float y = r2[lane + 32 * j]; v2[j] = y; s2 += y * y;
    }
#pragma unroll
    for (int off = 16; off; off >>= 1) {
        s1 += __shfl_xor(s1, off, 32);
        s2 += __shfl_xor(s2, off, 32);
    }
    const float c1 = w1 / fmaxf(sqrtf(s1), 1e-12f);   // w1 * invnorm(p1[i1])
    const float c2 = w2 / fmaxf(sqrtf(s2), 1e-12f);
    const float dn = 1.f / (w1 + w2 + 1e-6f);

    float f[16];
    float sf = 0.f;
#pragma unroll
    for (int j = 0; j < 16; ++j) {
        f[j] = (c1 * v1[j] + c2 * v2[j]) * dn;
        sf += f[j] * f[j];
    }
#pragma unroll
    for (int off = 16; off; off >>= 1) sf += __shfl_xor(sf, off, 32);
    const float invf = 1.f / fmaxf(sqrtf(sf), 1e-12f);
    float* dst = p_s_out + (size_t)t * DD;
#pragma unroll
    for (int j = 0; j < 16; ++j) dst[lane + 32 * j] = f[j] * invf;
}

// ---------------------------------------------------------------------------
extern "C" void kernel_launch(void* const* d_in, const int* in_sizes, int n_in,
                              void* d_out, int out_size, void* d_ws, size_t ws_size,
                              hipStream_t stream) {
    (void)in_sizes; (void)n_in; (void)out_size; (void)ws_size;
    const float* anchor = (const float*)d_in[0];
    const float* p1     = (const float*)d_in[1];
    const float* p2     = (const float*)d_in[2];
    const int*   idx_a  = (const int*)d_in[3];
    const int*   idx_p  = (const int*)d_in[4];

    float* A_s = (float*)d_out;                            // (B,K,D)
    float* P_s = (float*)d_out + (size_t)B_ * KK * DD;     // (B,K,D)

    // workspace: [ best (2*B*K u64) | Asel (B*K*D f32) ]
    unsigned long long* best = (unsigned long long*)d_ws;
    float* asel = (float*)((char*)d_ws + (size_t)2 * B_ * KK * sizeof(unsigned long long));

    init_best<<<dim3((2 * B_ * KK) / 256), 256, 0, stream>>>(best);
    prep_kernel<<<dim3((2 * B_ * KK) / 8), 256, 0, stream>>>(anchor, idx_a, idx_p, A_s, asel);

    dim3 sg(NP / 128, B_);
    score_kernel<<<sg, 256, 0, stream>>>(p1, asel, best);
    score_kernel<<<sg, 256, 0, stream>>>(p2, asel, best + B_ * KK);

    fuse_kernel<<<dim3((B_ * KK) / 8), 256, 0, stream>>>(p1, p2, best, P_s);
}